// SelfDescriptionCreator_27238682591399
// MI455X (gfx1250) — compile-verified
//
#include <hip/hip_runtime.h>
#include <hip/hip_bf16.h>

// ---------------- workspace layout (float element offsets) ----------------
#define WS_KL   0
#define WS_KERN 16                       // 4 levels * 27 floats, [l][c*9+tap]
#define WS_PART 128                      // 512 block partials
#define WS_TE   1024                     // 32*512*512 total_error
#define N_TE    (32L*512*512)
#define WS_DS2  (WS_TE + N_TE)           // 32*256*256*3
#define WS_DS3  (WS_DS2 + 32L*256*256*3) // 32*128*128*3
#define WS_DS4  (WS_DS3 + 32L*128*128*3) // 32*64*64*3
#define WS_TMP  (WS_DS4 + 32L*64*64*3)   // horizontal-pass temp (max 32*512*256*3)

typedef float f32x2 __attribute__((ext_vector_type(2)));
typedef float f32x4 __attribute__((ext_vector_type(4)));
typedef float f32x8 __attribute__((ext_vector_type(8)));

// ---------------- prep: softplus, sampled kernels, KL ----------------
__global__ void prep_kernel(const float* __restrict__ loc,
                            const float* __restrict__ scale,
                            const float* __restrict__ eps,
                            float* __restrict__ ws) {
  if (threadIdx.x != 0) return;
  float sig[27];
  float kl = 0.f;
  for (int i = 0; i < 27; ++i) {
    float s  = scale[i];
    float sg = log1pf(expf(s));          // softplus
    sig[i]   = sg;
    float mu = loc[i];
    kl += -logf(sg) + 0.5f * (sg * sg + mu * mu) - 0.5f;
  }
  ws[WS_KL] = kl;
  for (int l = 0; l < 4; ++l) {
    float v[27], sum[3] = {0.f, 0.f, 0.f};
    for (int t = 0; t < 9; ++t)
      for (int c = 0; c < 3; ++c) {
        int i = t * 3 + c;
        float val = (t == 4) ? 0.f : (loc[i] + sig[i] * eps[l * 27 + i]); // center masked
        v[i] = val;
        sum[c] += val;
      }
    for (int t = 0; t < 9; ++t)
      for (int c = 0; c < 3; ++c)
        ws[WS_KERN + l * 27 + c * 9 + t] = v[t * 3 + c] / (sum[c] + 1e-7f);
  }
}

// ------------- separable antialiased downsample (triangle, 2f taps) -------------
__global__ void down_h(const float* __restrict__ in, float* __restrict__ out,
                       int Win, int Wo, int f, long n) {
  long i = (long)blockIdx.x * blockDim.x + threadIdx.x;
  if (i >= n) return;
  int c = (int)(i % 3); long r = i / 3;
  int xo = (int)(r % Wo); long by = r / Wo;      // by = b*H + y
  float cc = f * xo + 0.5f * f - 0.5f;
  int   j0 = f * xo - f / 2;
  float s = 0.f, wsum = 0.f;
  for (int t = 0; t < 2 * f; ++t) {
    int j = j0 + t;
    float w = 1.f - fabsf((float)j - cc) / (float)f;
    if (j >= 0 && j < Win) { s += w * in[(by * Win + j) * 3 + c]; wsum += w; }
  }
  out[i] = s / wsum;
}

__global__ void down_v(const float* __restrict__ in, float* __restrict__ out,
                       int Hin, int Ho, int Wo, int f, long n) {
  long i = (long)blockIdx.x * blockDim.x + threadIdx.x;
  if (i >= n) return;
  int c = (int)(i % 3); long r = i / 3;
  int xo = (int)(r % Wo); r /= Wo;
  int yo = (int)(r % Ho); int b = (int)(r / Ho);
  float cc = f * yo + 0.5f * f - 0.5f;
  int   j0 = f * yo - f / 2;
  float s = 0.f, wsum = 0.f;
  for (int t = 0; t < 2 * f; ++t) {
    int j = j0 + t;
    float w = 1.f - fabsf((float)j - cc) / (float)f;
    if (j >= 0 && j < Hin) { s += w * in[(((long)b * Hin + j) * Wo + xo) * 3 + c]; wsum += w; }
  }
  out[i] = s / wsum;
}

// ------------- fused: conv err tile in LDS (async-loaded) + upsample accumulate -------------
#define TS 32
__global__ __launch_bounds__(256) void level_accum(
    const float* __restrict__ ds, const float* __restrict__ kerng,
    float* __restrict__ te, int f, int th, int tw, int writeMode) {
  __shared__ float kco[27];
  __shared__ float dst[36 * 36 * 3];   // ds tile (zero-padded outside image)
  __shared__ float ert[34 * 34 * 3];   // err tile

  int tid = threadIdx.x;
  if (tid < 27) kco[tid] = kerng[tid];

  int x0 = blockIdx.x * TS, y0 = blockIdx.y * TS, b = blockIdx.z;
  int e0x = (f == 1) ? x0 : (x0 / f - 1);
  int e0y = (f == 1) ? y0 : (y0 / f - 1);
  int errW = TS / f + 2;
  int dsW  = errW + 2;
  int d0x = e0x - 1, d0y = e0y - 1;

  int dsN = dsW * dsW * 3;
  for (int k = tid; k < dsN; k += 256) dst[k] = 0.f;   // SAME-pad zeros
  __syncthreads();

  // CDNA5 async global->LDS copy of the in-range tile, tracked by ASYNCcnt
  const float* dsb = ds + (long)b * th * tw * 3;
  for (int k = tid; k < dsN; k += 256) {
    int c = k % 3; int q = k / 3;
    int kx = q % dsW, ky = q / dsW;
    int gy = d0y + ky, gx = d0x + kx;
    if (gy >= 0 && gy < th && gx >= 0 && gx < tw) {
      unsigned laddr = (unsigned)(unsigned long long)(&dst[k]);
      unsigned long long gaddr =
          (unsigned long long)(const void*)(dsb + ((long)gy * tw + gx) * 3 + c);
      asm volatile("global_load_async_to_lds_b32 %0, %1, off"
                   :: "v"(laddr), "v"(gaddr) : "memory");
    }
  }
#if __has_builtin(__builtin_amdgcn_s_wait_asynccnt)
  __builtin_amdgcn_s_wait_asynccnt(0);
#else
  asm volatile("s_wait_asynccnt 0" ::: "memory");
#endif
  __syncthreads();

  // err = ds - depthwise3x3(ds), computed once per level pixel (clamped slots)
  int eN = errW * errW * 3;
  for (int k = tid; k < eN; k += 256) {
    int c = k % 3; int q = k / 3;
    int ex = q % errW, ey = q / errW;
    int py = min(max(e0y + ey, 0), th - 1);
    int px = min(max(e0x + ex, 0), tw - 1);
    int ty = py - d0y, tx = px - d0x;
    float conv = 0.f;
#pragma unroll
    for (int dy = 0; dy < 3; ++dy)
#pragma unroll
      for (int dx = 0; dx < 3; ++dx)
        conv += kco[c * 9 + dy * 3 + dx] *
                dst[((ty - 1 + dy) * dsW + (tx - 1 + dx)) * 3 + c];
    ert[(ey * errW + ex) * 3 + c] = dst[(ty * dsW + tx) * 3 + c] - conv;
  }
  __syncthreads();

  // bilinear upsample (no antialias; edge renorm == clamp) + channel sum
  float inv_f = 1.0f / (float)f;
  for (int q = tid; q < TS * TS; q += 256) {
    int ox = x0 + q % TS, oy = y0 + q / TS;
    float sx = (ox + 0.5f) * inv_f - 0.5f;
    float sy = (oy + 0.5f) * inv_f - 0.5f;
    int ix = (int)floorf(sx), iy = (int)floorf(sy);
    float wx = sx - ix, wy = sy - iy;
    int kx0 = min(max(ix, 0), tw - 1) - e0x;
    int kx1 = min(max(ix + 1, 0), tw - 1) - e0x;
    int ky0 = min(max(iy, 0), th - 1) - e0y;
    int ky1 = min(max(iy + 1, 0), th - 1) - e0y;
    float val = 0.f;
#pragma unroll
    for (int c = 0; c < 3; ++c) {
      float e00 = ert[(ky0 * errW + kx0) * 3 + c];
      float e01 = ert[(ky0 * errW + kx1) * 3 + c];
      float e10 = ert[(ky1 * errW + kx0) * 3 + c];
      float e11 = ert[(ky1 * errW + kx1) * 3 + c];
      val += (1.f - wy) * ((1.f - wx) * e00 + wx * e01) +
             wy * ((1.f - wx) * e10 + wx * e11);
    }
    long o = ((long)b * 512 + oy) * 512 + ox;
    if (writeMode) te[o] = val; else te[o] += val;
  }
}

// ------------- sum-of-squares reduction via V_WMMA_F32_16X16X4_F32 -------------
// D += sq(A) x ones : every column of D is identical, full-D sum = 16 * true sum.
// b128 loads: 16 B/lane/iter (512 B/wave), two WMMAs per load.
__global__ __launch_bounds__(256) void reduce_sq_wmma(const float* __restrict__ te,
                                                      float* __restrict__ part, long n) {
  __shared__ float wsum[8];
  int tid = threadIdx.x, lane = tid & 31, wv = tid >> 5;
  long nwaves = (long)gridDim.x * 8;
  long gw = (long)blockIdx.x * 8 + wv;
  long per = n / nwaves;                 // 2048 with 512x256 over 8.4M
  long base = gw * per;
  f32x8 acc = {0.f, 0.f, 0.f, 0.f, 0.f, 0.f, 0.f, 0.f};
  f32x2 ones; ones[0] = 1.f; ones[1] = 1.f;
  for (long i = 0; i < per; i += 128) {  // 128 elems per wave per iter
    f32x4 v = *(const f32x4*)(te + base + i + 4 * lane);  // 16B aligned
    f32x2 a0; a0[0] = v[0] * v[0]; a0[1] = v[1] * v[1];
    f32x2 a1; a1[0] = v[2] * v[2]; a1[1] = v[3] * v[3];
    acc = __builtin_amdgcn_wmma_f32_16x16x4_f32(
        false, a0, false, ones, (short)0, acc, false, false);
    acc = __builtin_amdgcn_wmma_f32_16x16x4_f32(
        false, a1, false, ones, (short)0, acc, false, false);
  }
  float s = 0.f;
#pragma unroll
  for (int j = 0; j < 8; ++j) s += acc[j];
  for (int off = 16; off; off >>= 1) s += __shfl_xor(s, off, 32);
  if (lane == 0) wsum[wv] = s;
  __syncthreads();
  if (tid == 0) {
    float t = 0.f;
    for (int j = 0; j < 8; ++j) t += wsum[j];
    part[blockIdx.x] = t;
  }
}

__global__ void finalize_kernel(const float* __restrict__ ws, float* __restrict__ out) {
  __shared__ float red[256];
  int tid = threadIdx.x;
  red[tid] = ws[WS_PART + tid] + ws[WS_PART + 256 + tid];
  __syncthreads();
  for (int o = 128; o; o >>= 1) {
    if (tid < o) red[tid] += red[tid + o];
    __syncthreads();
  }
  if (tid == 0)
    out[0] = red[0] / (16.0f * 32.0f) + 4.0f * ws[WS_KL];  // recon/B + L*KL (16x WMMA factor)
}

// ---------------- host launch ----------------
extern "C" void kernel_launch(void* const* d_in, const int* in_sizes, int n_in,
                              void* d_out, int out_size, void* d_ws, size_t ws_size,
                              hipStream_t stream) {
  const float* inp = (const float*)d_in[0];
  const float* loc = (const float*)d_in[1];
  const float* scl = (const float*)d_in[2];
  const float* eps = (const float*)d_in[3];
  float* ws  = (float*)d_ws;
  float* out = (float*)d_out;

  float* te  = ws + WS_TE;
  float* ds2 = ws + WS_DS2;
  float* ds3 = ws + WS_DS3;
  float* ds4 = ws + WS_DS4;
  float* tmp = ws + WS_TMP;

  prep_kernel<<<1, 32, 0, stream>>>(loc, scl, eps, ws);

  { long n = 32L * 512 * 256 * 3;
    down_h<<<(int)((n + 255) / 256), 256, 0, stream>>>(inp, tmp, 512, 256, 2, n);
    long m = 32L * 256 * 256 * 3;
    down_v<<<(int)((m + 255) / 256), 256, 0, stream>>>(tmp, ds2, 512, 256, 256, 2, m); }
  { long n = 32L * 512 * 128 * 3;
    down_h<<<(int)((n + 255) / 256), 256, 0, stream>>>(inp, tmp, 512, 128, 4, n);
    long m = 32L * 128 * 128 * 3;
    down_v<<<(int)((m + 255) / 256), 256, 0, stream>>>(tmp, ds3, 512, 128, 128, 4, m); }
  { long n = 32L * 512 * 64 * 3;
    down_h<<<(int)((n + 255) / 256), 256, 0, stream>>>(inp, tmp, 512, 64, 8, n);
    long m = 32L * 64 * 64 * 3;
    down_v<<<(int)((m + 255) / 256), 256, 0, stream>>>(tmp, ds4, 512, 64, 64, 8, m); }

  dim3 g(512 / TS, 512 / TS, 32);
  level_accum<<<g, 256, 0, stream>>>(inp, ws + WS_KERN +  0, te, 1, 512, 512, 1);
  level_accum<<<g, 256, 0, stream>>>(ds2, ws + WS_KERN + 27, te, 2, 256, 256, 0);
  level_accum<<<g, 256, 0, stream>>>(ds3, ws + WS_KERN + 54, te, 4, 128, 128, 0);
  level_accum<<<g, 256, 0, stream>>>(ds4, ws + WS_KERN + 81, te, 8,  64,  64, 0);

  reduce_sq_wmma<<<512, 256, 0, stream>>>(te, ws + WS_PART, N_TE);
  finalize_kernel<<<1, 256, 0, stream>>>(ws, out);
}